// GATBlock_31301721653958
// MI455X (gfx1250) — compile-verified
//
#include <hip/hip_runtime.h>
#include <hip/hip_bf16.h>

// ---------------------------------------------------------------------------
// GATv2 block for MI455X (gfx1250, wave32).
// Algebraic optimization: edge features depend only on relation id, so
// e = (relations @ W_e)[relation_index]  and  loop_e = sum_r count[r]*rel_e[r]/E.
// GEMMs use v_wmma_f32_16x16x4_f32 (one 16x16 tile per wave32); the x_l/x_r
// transforms are fused into one dual-accumulator WMMA kernel sharing A loads.
// ---------------------------------------------------------------------------

typedef __attribute__((ext_vector_type(2))) float v2f;
typedef __attribute__((ext_vector_type(8))) float v8f;

#define HC 256   // H*C
#define CD 128   // C (= out dim, also D)

// ---- monotone float<->uint encoding for atomicMax on floats ---------------
__device__ __forceinline__ unsigned f2ord(float f) {
  unsigned u = __float_as_uint(f);
  return (u & 0x80000000u) ? ~u : (u | 0x80000000u);
}
__device__ __forceinline__ float ord2f(unsigned k) {
  unsigned u = (k & 0x80000000u) ? (k ^ 0x80000000u) : ~k;
  return __uint_as_float(u);
}
#define ORD_NEG_INF 0x007FFFFFu   // f2ord(-inf)

// ---------------------------------------------------------------------------
// Fused dual GEMM (shared A): Out0 = A@B0 + bias0, Out1 = A@B1 + bias1.
// One wave computes the same 16x16 tile of both outputs; A loaded once.
// ---------------------------------------------------------------------------
__global__ void gemm_dual_wmma_f32(const float* __restrict__ A,
                                   const float* __restrict__ B0,
                                   const float* __restrict__ bias0,
                                   float* __restrict__ Out0,
                                   const float* __restrict__ B1,
                                   const float* __restrict__ bias1,
                                   float* __restrict__ Out1,
                                   int M, int K, int Nc) {
  int wave  = (int)((blockIdx.x * blockDim.x + threadIdx.x) >> 5);
  int lane  = (int)(threadIdx.x & 31);
  int tilesN = Nc >> 4;
  int tilesM = M >> 4;
  if (wave >= tilesM * tilesN) return;
  int m0 = (wave / tilesN) << 4;
  int n0 = (wave % tilesN) << 4;
  int half = lane >> 4;
  int l15  = lane & 15;

  const float* Arow  = A  + (size_t)(m0 + l15) * K + 2 * half;
  const float* Bcol0 = B0 + (size_t)(2 * half) * Nc + n0 + l15;
  const float* Bcol1 = B1 + (size_t)(2 * half) * Nc + n0 + l15;
  v8f acc0 = {}, acc1 = {};
  #pragma unroll 4
  for (int k0 = 0; k0 < K; k0 += 4) {
    v2f av; av[0] = Arow[k0]; av[1] = Arow[k0 + 1];
    v2f bv0; bv0[0] = Bcol0[(size_t)k0 * Nc]; bv0[1] = Bcol0[(size_t)k0 * Nc + Nc];
    v2f bv1; bv1[0] = Bcol1[(size_t)k0 * Nc]; bv1[1] = Bcol1[(size_t)k0 * Nc + Nc];
    acc0 = __builtin_amdgcn_wmma_f32_16x16x4_f32(false, av, false, bv0,
                                                 (short)0, acc0, false, false);
    acc1 = __builtin_amdgcn_wmma_f32_16x16x4_f32(false, av, false, bv1,
                                                 (short)0, acc1, false, false);
  }
  float bb0 = bias0 ? bias0[n0 + l15] : 0.f;
  float bb1 = bias1 ? bias1[n0 + l15] : 0.f;
  #pragma unroll
  for (int i = 0; i < 8; ++i) {
    size_t row = (size_t)(m0 + half * 8 + i);
    Out0[row * Nc + n0 + l15] = acc0[i] + bb0;
    Out1[row * Nc + n0 + l15] = acc1[i] + bb1;
  }
}

// ---------------------------------------------------------------------------
// Generic fp32 WMMA GEMM: Out[M,Nc] = (reluA ? relu(A) : A)[M,K] @ B[K,Nc] + bias
// ---------------------------------------------------------------------------
__global__ void gemm_wmma_f32(const float* __restrict__ A,
                              const float* __restrict__ B,
                              const float* __restrict__ bias,
                              float* __restrict__ Out,
                              int M, int K, int Nc, int reluA) {
  int wave  = (int)((blockIdx.x * blockDim.x + threadIdx.x) >> 5);
  int lane  = (int)(threadIdx.x & 31);
  int tilesN = Nc >> 4;
  int tilesM = M >> 4;
  if (wave >= tilesM * tilesN) return;
  int m0 = (wave / tilesN) << 4;
  int n0 = (wave % tilesN) << 4;
  int half = lane >> 4;
  int l15  = lane & 15;

  const float* Arow = A + (size_t)(m0 + l15) * K + 2 * half;
  const float* Bcol = B + (size_t)(2 * half) * Nc + n0 + l15;
  v8f acc = {};
  #pragma unroll 4
  for (int k0 = 0; k0 < K; k0 += 4) {
    float a0 = Arow[k0];
    float a1 = Arow[k0 + 1];
    if (reluA) { a0 = fmaxf(a0, 0.f); a1 = fmaxf(a1, 0.f); }
    v2f av; av[0] = a0; av[1] = a1;
    v2f bv; bv[0] = Bcol[(size_t)k0 * Nc]; bv[1] = Bcol[(size_t)k0 * Nc + Nc];
    acc = __builtin_amdgcn_wmma_f32_16x16x4_f32(false, av, false, bv,
                                                (short)0, acc, false, false);
  }
  float bb = bias ? bias[n0 + l15] : 0.f;
  #pragma unroll
  for (int i = 0; i < 8; ++i)
    Out[(size_t)(m0 + half * 8 + i) * Nc + n0 + l15] = acc[i] + bb;
}

// ---------------------------------------------------------------------------
// Init: zero node output region, counts, denominators; -inf the max buffer.
// ---------------------------------------------------------------------------
__global__ void init_kernel(float* __restrict__ node_out, unsigned* __restrict__ nmax,
                            float* __restrict__ den, int* __restrict__ counts, int N) {
  int i = (int)(blockIdx.x * blockDim.x + threadIdx.x);
  if (i < N * CD) node_out[i] = 0.f;
  if (i < N * 2) { nmax[i] = ORD_NEG_INF; den[i] = 0.f; }
  if (i < 1024) counts[i] = 0;
}

// ---------------------------------------------------------------------------
// Relation-id histogram (LDS-staged).
// ---------------------------------------------------------------------------
__global__ void hist_kernel(const int* __restrict__ relidx, int* __restrict__ counts,
                            int E, int R) {
  __shared__ int h[1024];
  for (int i = threadIdx.x; i < R; i += blockDim.x) h[i] = 0;
  __syncthreads();
  int idx = (int)(blockIdx.x * blockDim.x + threadIdx.x);
  int stride = (int)(gridDim.x * blockDim.x);
  for (int j = idx; j < E; j += stride) atomicAdd(&h[relidx[j]], 1);
  __syncthreads();
  for (int i = threadIdx.x; i < R; i += blockDim.x)
    if (h[i]) atomicAdd(&counts[i], h[i]);
}

// loop_e[c] = (1/E) * sum_r count[r] * rel_e[r][c]   (one block, 256 threads)
__global__ void loope_kernel(const int* __restrict__ counts, const float* __restrict__ rel_e,
                             float* __restrict__ loop_e, int R, int E) {
  int c = threadIdx.x;
  float s = 0.f;
  for (int r = 0; r < R; ++r) s += (float)counts[r] * rel_e[(size_t)r * HC + c];
  loop_e[c] = s / (float)E;
}

// ---------------------------------------------------------------------------
// Pass 1: per-edge attention logits (one wave per edge; lanes 0-15 head 0,
// lanes 16-31 head 1), atomic running max per (dst, head).
// ---------------------------------------------------------------------------
__global__ void edge_logits_kernel(const int* __restrict__ src, const int* __restrict__ dst,
                                   const int* __restrict__ relidx,
                                   const float* __restrict__ xl, const float* __restrict__ xr,
                                   const float* __restrict__ rel_e, const float* __restrict__ loop_e,
                                   const float* __restrict__ att,
                                   float* __restrict__ logits, unsigned* __restrict__ nmax,
                                   int E, int N) {
  int wid  = (int)((blockIdx.x * blockDim.x + threadIdx.x) >> 5);
  int lane = (int)(threadIdx.x & 31);
  if (wid >= E + N) return;
  int s, d;
  const float* ep;
  if (wid < E) {
    s = src[wid]; d = dst[wid];
    ep = rel_e + (size_t)relidx[wid] * HC;
  } else {
    s = d = wid - E;
    ep = loop_e;
  }
  int c0 = lane * 8;  // 8 contiguous channels per lane
  const float4* A4 = reinterpret_cast<const float4*>(xl + (size_t)s * HC + c0);
  const float4* B4 = reinterpret_cast<const float4*>(xr + (size_t)d * HC + c0);
  const float4* E4 = reinterpret_cast<const float4*>(ep + c0);
  const float4* T4 = reinterpret_cast<const float4*>(att + c0);
  float p = 0.f;
  #pragma unroll
  for (int t = 0; t < 2; ++t) {
    float4 a = A4[t], b = B4[t], e = E4[t], w = T4[t];
    float v;
    v = a.x + b.x + e.x; v = v > 0.f ? v : 0.2f * v; p += v * w.x;
    v = a.y + b.y + e.y; v = v > 0.f ? v : 0.2f * v; p += v * w.y;
    v = a.z + b.z + e.z; v = v > 0.f ? v : 0.2f * v; p += v * w.z;
    v = a.w + b.w + e.w; v = v > 0.f ? v : 0.2f * v; p += v * w.w;
  }
  // butterfly reduce within each 16-lane half (one head per half)
  for (int o = 8; o >= 1; o >>= 1) p += __shfl_xor(p, o, 32);
  if ((lane & 15) == 0) {
    int h = lane >> 4;
    logits[(size_t)wid * 2 + h] = p;
    atomicMax(&nmax[(size_t)d * 2 + h], f2ord(p));
  }
}

// Pass 2: ex = exp(logit - max[d]); accumulate denominator. (in-place on logits)
__global__ void edge_exp_kernel(const int* __restrict__ dst, float* __restrict__ logits,
                                const unsigned* __restrict__ nmax, float* __restrict__ den,
                                int E, int N) {
  int idx = (int)(blockIdx.x * blockDim.x + threadIdx.x);
  if (idx >= (E + N) * 2) return;
  int j = idx >> 1, h = idx & 1;
  int d = (j < E) ? dst[j] : (j - E);
  float mx = ord2f(nmax[(size_t)d * 2 + h]);
  float ex = expf(logits[idx] - mx);
  logits[idx] = ex;
  atomicAdd(&den[(size_t)d * 2 + h], ex);
}

// Pass 3: node_out[d][c] += 0.5 * alpha[j,h] * x_l[s][h*C+c]  (head-mean fused)
__global__ void edge_scatter_kernel(const int* __restrict__ src, const int* __restrict__ dst,
                                    const float* __restrict__ xl, const float* __restrict__ exbuf,
                                    const float* __restrict__ den, float* __restrict__ node_out,
                                    int E, int N) {
  int j = (int)blockIdx.x;          // one edge per block
  int c = (int)threadIdx.x;         // 0..255 over H*C
  int h = c >> 7;
  int s, d;
  if (j < E) { s = src[j]; d = dst[j]; } else { s = d = j - E; }
  float alpha = exbuf[(size_t)j * 2 + h] / den[(size_t)d * 2 + h];
  float val = 0.5f * alpha * xl[(size_t)s * HC + c];
  atomicAdd(&node_out[(size_t)d * CD + (c & (CD - 1))], val);
}

// Final node bias.
__global__ void bias_kernel(float* __restrict__ node_out, const float* __restrict__ bias, int N) {
  int i = (int)(blockIdx.x * blockDim.x + threadIdx.x);
  if (i < N * CD) node_out[i] += bias[i & (CD - 1)];
}

// ---------------------------------------------------------------------------
extern "C" void kernel_launch(void* const* d_in, const int* in_sizes, int n_in,
                              void* d_out, int out_size, void* d_ws, size_t ws_size,
                              hipStream_t stream) {
  const float* x          = (const float*)d_in[0];
  const int*   edge_index = (const int*)  d_in[1];
  const float* relations  = (const float*)d_in[2];
  const int*   relidx     = (const int*)  d_in[3];
  const float* W_l        = (const float*)d_in[4];
  const float* b_l        = (const float*)d_in[5];
  const float* W_r        = (const float*)d_in[6];
  const float* b_r        = (const float*)d_in[7];
  const float* W_e        = (const float*)d_in[8];
  const float* att        = (const float*)d_in[9];
  const float* bias       = (const float*)d_in[10];
  const float* W_lin2     = (const float*)d_in[11];
  const float* b_lin2     = (const float*)d_in[12];

  const int D = 128;
  const int E = in_sizes[3];          // 500000
  const int N = in_sizes[0] / D;      // 50000
  const int R = in_sizes[2] / D;      // 1024

  const int* src = edge_index;
  const int* dst = edge_index + E;

  float* node_out = (float*)d_out;                 // [N, 128]
  float* rel_out  = node_out + (size_t)N * CD;     // [R, 128]

  // workspace carve-out (256B aligned slices)
  char* ws = (char*)d_ws;
  size_t off = 0;
  auto carve = [&](size_t bytes) -> void* {
    void* p = ws + off;
    off += (bytes + 255) & ~(size_t)255;
    return p;
  };
  float*    x_l    = (float*)   carve((size_t)N * HC * 4);     // 51.2 MB
  float*    x_r    = (float*)   carve((size_t)N * HC * 4);     // 51.2 MB
  float*    rel_e  = (float*)   carve((size_t)R * HC * 4);     //  1.0 MB
  float*    loop_e = (float*)   carve((size_t)HC * 4);
  float*    logits = (float*)   carve((size_t)(E + N) * 2 * 4);//  4.4 MB
  unsigned* nmax   = (unsigned*)carve((size_t)N * 2 * 4);
  float*    den    = (float*)   carve((size_t)N * 2 * 4);
  int*      counts = (int*)     carve((size_t)R * 4);
  (void)ws_size; (void)n_in; (void)out_size;

  const int TPB = 256;

  // 0) init accumulators / output region
  init_kernel<<<(N * CD + TPB - 1) / TPB, TPB, 0, stream>>>(node_out, nmax, den, counts, N);

  // 1) WMMA GEMMs: fused x_l/x_r (shared A), then rel_e = relations@W_e
  {
    int tiles = (N / 16) * (HC / 16);
    int blocks = (tiles * 32 + TPB - 1) / TPB;
    gemm_dual_wmma_f32<<<blocks, TPB, 0, stream>>>(x, W_l, b_l, x_l,
                                                   W_r, b_r, x_r, N, D, HC);
  }
  {
    int tiles = (R / 16) * (HC / 16);
    gemm_wmma_f32<<<(tiles * 32 + TPB - 1) / TPB, TPB, 0, stream>>>(
        relations, W_e, nullptr, rel_e, R, D, HC, 0);
  }

  // 2) relation histogram -> self-loop edge feature in transformed space
  hist_kernel<<<128, TPB, 0, stream>>>(relidx, counts, E, R);
  loope_kernel<<<1, HC, 0, stream>>>(counts, rel_e, loop_e, R, E);

  // 3) attention passes
  {
    int waves = E + N;
    edge_logits_kernel<<<(waves * 32 + TPB - 1) / TPB, TPB, 0, stream>>>(
        src, dst, relidx, x_l, x_r, rel_e, loop_e, att, logits, nmax, E, N);
  }
  {
    int total = (E + N) * 2;
    edge_exp_kernel<<<(total + TPB - 1) / TPB, TPB, 0, stream>>>(dst, logits, nmax, den, E, N);
  }
  edge_scatter_kernel<<<E + N, HC, 0, stream>>>(src, dst, x_l, logits, den, node_out, E, N);
  bias_kernel<<<(N * CD + TPB - 1) / TPB, TPB, 0, stream>>>(node_out, bias, N);

  // 4) relation branch: rel_out = relu(rel_e) @ W_lin2 + b_lin2  (WMMA, relu on A)
  {
    int tiles = (R / 16) * (CD / 16);
    gemm_wmma_f32<<<(tiles * 32 + TPB - 1) / TPB, TPB, 0, stream>>>(
        rel_e, W_lin2, b_lin2, rel_out, R, HC, CD, 1);
  }
}